// AttentionFilter_10514079940621
// MI455X (gfx1250) — compile-verified
//
#include <hip/hip_runtime.h>
#include <math.h>

// ---------------------------------------------------------------------------
// AttentionFilter on MI455X (gfx1250): whole pipeline as fp32 WMMA GEMMs.
//   B=2, C=128, H=256, W=510, WF=256.  All DFT/iDFT stages are matmuls so the
//   ~257 GFLOP of work runs on V_WMMA_F32_16X16X4_F32 instead of the VALU.
// Round-2 refinements:
//   * B-side LDS tiles stored transposed [n][k] with pad-to-18 stride so every
//     A and B fragment is one aligned 8-byte LDS load (ds_load_b64 straight
//     into an even VGPR pair -> no v_mov packing, fewer ds ops per WMMA).
//   * global_prefetch of the next k-tile overlaps HBM/L2 fill with WMMA work.
// ---------------------------------------------------------------------------

typedef __attribute__((ext_vector_type(2))) float v2f;
typedef __attribute__((ext_vector_type(8))) float v8f;

#define TM 64
#define TN 64
#define TK 16
#define PK 18  // padded LDS row stride (even -> 8B-aligned pairs, spreads banks)

static __device__ __forceinline__ v8f wmma4(v2f a, v2f b, v8f c) {
  // D = A(16x4) * B(4x16) + C(16x16), all fp32. NEG/C-mod left at 0 (fp32 NEG
  // bits are reserved except CNeg per ISA 7.12), signs are applied in VALU.
  return __builtin_amdgcn_wmma_f32_16x16x4_f32(false, a, false, b, (short)0, c,
                                               false, false);
}

static __device__ __forceinline__ v2f lds_pair(const float* p) {
  return *reinterpret_cast<const v2f*>(p);
}

#define NB 2
#define NC 128
#define NH 256
#define NW 510
#define NWF 256
#define NHW (NH * NW)          // 130560
#define LD 512                 // complex row layout: [re 0..255 | im 256..511]
#define MROWS (NB * NC * NH)   // 65536

// ---------------------------------------------------------------------------
// DFT basis tables (phase reduced with exact integer modulo before cosf/sinf).
// ---------------------------------------------------------------------------
__global__ void init_dftW(float* __restrict__ t) {  // [510][512], ortho 1/sqrt(W)
  int i = blockIdx.x * 256 + threadIdx.x;
  if (i >= NW * LD) return;
  int w = i / LD, k = i % LD, kk = k & 255;
  int ph = (w * kk) % NW;
  float th = 6.283185307179586f * (float)ph / (float)NW;
  float sc = rsqrtf((float)NW);
  t[i] = (k < 256) ? cosf(th) * sc : -sinf(th) * sc;
}

__global__ void init_dftH(float* __restrict__ ct, float* __restrict__ st) {
  // cos/sin [256][256], scaled by 1/sqrt(H)=1/16 (ortho, both directions)
  int i = blockIdx.x * 256 + threadIdx.x;
  if (i >= NH * NH) return;
  int r = i >> 8, c = i & 255;
  int ph = (r * c) & 255;
  float th = 6.283185307179586f * (float)ph / 256.0f;
  ct[i] = cosf(th) * 0.0625f;
  st[i] = sinf(th) * 0.0625f;
}

__global__ void init_irW(float* __restrict__ t) {  // [512][510] irfft basis
  int i = blockIdx.x * 256 + threadIdx.x;
  if (i >= LD * NW) return;
  int r = i / NW, w = i % NW, k = r & 255;
  int ph = (k * w) % NW;
  float th = 6.283185307179586f * (float)ph / (float)NW;
  float wk = (k == 0 || k == 255) ? 1.0f : 2.0f;  // k=255 is Nyquist (W=510)
  float sc = wk * rsqrtf((float)NW);
  t[i] = (r < 256) ? cosf(th) * sc : -sinf(th) * sc;
}

// ---------------------------------------------------------------------------
// Generic batched fp32 WMMA GEMM:  Cmat = A[MxK] @ Bmat[KxN] (+ bias per row)
// 64x64 macro tile, 128 threads = 4 waves, each wave a 32x32 tile (2x2 frags).
// LDS: A as [m][k] (stride PK), B transposed as [n][k] (stride PK).
// ---------------------------------------------------------------------------
__global__ __launch_bounds__(128) void gemm_f32_kernel(
    const float* __restrict__ A, int lda, long strideA,
    const float* __restrict__ Bm, int ldb, long strideB,
    float* __restrict__ Cm, int ldc, long strideC,
    int M, int N, int K, const float* __restrict__ bias) {
  __shared__ float lA[TM * PK];
  __shared__ float lB[TN * PK];
  int bz = blockIdx.z;
  A += (long)bz * strideA;
  Bm += (long)bz * strideB;
  Cm += (long)bz * strideC;
  int tileN = blockIdx.x * TN, tileM = blockIdx.y * TM;
  int tid = threadIdx.x, lane = tid & 31, wave = tid >> 5;
  int wm = (wave >> 1) * 32, wn = (wave & 1) * 32;
  int lm = lane & 15, klo = (lane >> 4) * 2;  // ISA 7.12.2 fp32 A/B layout
  v8f acc[2][2] = {};
  for (int k0 = 0; k0 < K; k0 += TK) {
    for (int i = tid; i < TM * TK; i += 128) {
      int r = i / TK, c = i % TK, gm = tileM + r, gk = k0 + c;
      lA[r * PK + c] = (gm < M && gk < K) ? A[(long)gm * lda + gk] : 0.0f;
    }
    for (int i = tid; i < TK * TN; i += 128) {
      int r = i / TN, c = i % TN, gk = k0 + r, gn = tileN + c;
      lB[c * PK + r] = (gk < K && gn < N) ? Bm[(long)gk * ldb + gn] : 0.0f;
    }
    // prefetch next k-tile while this one is consumed (global_prefetch_b8)
    if (k0 + TK < K) {
      int pm = tileM + (tid >> 1);
      if (pm < M) __builtin_prefetch(&A[(long)pm * lda + k0 + TK + (tid & 1) * 8], 0, 1);
      int pk = k0 + TK + (tid >> 4);
      int pn = tileN + (tid & 15) * 4;
      if (pk < K && pn < N) __builtin_prefetch(&Bm[(long)pk * ldb + pn], 0, 1);
    }
    __syncthreads();
    for (int kb = 0; kb < TK; kb += 4) {
      v2f a[2], b[2];
      for (int i = 0; i < 2; i++)
        a[i] = lds_pair(&lA[(wm + 16 * i + lm) * PK + kb + klo]);
      for (int j = 0; j < 2; j++)
        b[j] = lds_pair(&lB[(wn + 16 * j + lm) * PK + kb + klo]);
      for (int i = 0; i < 2; i++)
        for (int j = 0; j < 2; j++) acc[i][j] = wmma4(a[i], b[j], acc[i][j]);
    }
    __syncthreads();
  }
  int rhalf = (lane >> 4) * 8;  // C/D layout: row = vgpr + 8*(lane>=16)
  for (int i = 0; i < 2; i++)
    for (int j = 0; j < 2; j++) {
      int gn = tileN + wn + 16 * j + lm;
      if (gn >= N) continue;
      for (int r = 0; r < 8; r++) {
        int gm = tileM + wm + 16 * i + rhalf + r;
        if (gm < M) {
          float v = acc[i][j][r];
          if (bias) v += bias[gm];
          Cm[(long)gm * ldc + gn] = v;
        }
      }
    }
}

// ---------------------------------------------------------------------------
// Complex DFT along H per (b,c):  G[kh][kw] = sum_h exp(-+2*pi*i*kh*h/H)*F[h][kw]
// F/G layout per bc: [256 rows][re 0..255 | im 256..511].  Optional complex
// filter epilogue (GFNet filter [C][H][WF][2]).  4 WMMA accumulations / frag.
// ---------------------------------------------------------------------------
__global__ __launch_bounds__(128) void cdft_h_kernel(
    const float* __restrict__ F, float* __restrict__ G,
    const float* __restrict__ cosT, const float* __restrict__ sinT,
    const float* __restrict__ filt, int inverse) {
  __shared__ float lC[TM * PK], lS[TM * PK];
  __shared__ float lRe[TN * PK], lIm[TN * PK];  // transposed [kw][h]
  int bc = blockIdx.z;
  const float* Fb = F + (long)bc * NH * LD;
  float* Gb = G + (long)bc * NH * LD;
  int tileN = blockIdx.x * TN;  // kw
  int tileM = blockIdx.y * TM;  // kh
  int tid = threadIdx.x, lane = tid & 31, wave = tid >> 5;
  int wm = (wave >> 1) * 32, wn = (wave & 1) * 32;
  int lm = lane & 15, klo = (lane >> 4) * 2;
  v8f aRe[2][2] = {}, aIm[2][2] = {};
  for (int k0 = 0; k0 < NH; k0 += TK) {
    for (int i = tid; i < TM * TK; i += 128) {
      int r = i / TK, c = i % TK;
      lC[r * PK + c] = cosT[(long)(tileM + r) * NH + k0 + c];
      lS[r * PK + c] = sinT[(long)(tileM + r) * NH + k0 + c];
    }
    for (int i = tid; i < TK * TN; i += 128) {
      int r = i / TN, c = i % TN;
      lRe[c * PK + r] = Fb[(long)(k0 + r) * LD + tileN + c];
      lIm[c * PK + r] = Fb[(long)(k0 + r) * LD + 256 + tileN + c];
    }
    if (k0 + TK < NH) {
      int pk = k0 + TK + (tid >> 3);
      __builtin_prefetch(&Fb[(long)pk * LD + tileN + (tid & 7) * 8], 0, 1);
      __builtin_prefetch(&Fb[(long)pk * LD + 256 + tileN + (tid & 7) * 8], 0, 1);
    }
    __syncthreads();
    for (int kb = 0; kb < TK; kb += 4) {
      v2f ac[2], as_[2], bre[2], bim[2];
      for (int i = 0; i < 2; i++) {
        int m = (wm + 16 * i + lm) * PK + kb + klo;
        ac[i] = lds_pair(&lC[m]);
        as_[i] = lds_pair(&lS[m]);
      }
      for (int j = 0; j < 2; j++) {
        int n = (wn + 16 * j + lm) * PK + kb + klo;
        bre[j] = lds_pair(&lRe[n]);
        bim[j] = lds_pair(&lIm[n]);
      }
      for (int i = 0; i < 2; i++) {
        v2f ns; ns.x = -as_[i].x; ns.y = -as_[i].y;
        for (int j = 0; j < 2; j++) {
          aRe[i][j] = wmma4(ac[i], bre[j], aRe[i][j]);
          aIm[i][j] = wmma4(ac[i], bim[j], aIm[i][j]);
          if (!inverse) {  // e^{-i t}: Re += sin@Fim ; Im += -sin@Fre
            aRe[i][j] = wmma4(as_[i], bim[j], aRe[i][j]);
            aIm[i][j] = wmma4(ns, bre[j], aIm[i][j]);
          } else {         // e^{+i t}: Re += -sin@Fim ; Im += sin@Fre
            aRe[i][j] = wmma4(ns, bim[j], aRe[i][j]);
            aIm[i][j] = wmma4(as_[i], bre[j], aIm[i][j]);
          }
        }
      }
    }
    __syncthreads();
  }
  int ch = bc & (NC - 1), rhalf = (lane >> 4) * 8;
  for (int i = 0; i < 2; i++)
    for (int j = 0; j < 2; j++) {
      int gn = tileN + wn + 16 * j + lm;
      for (int r = 0; r < 8; r++) {
        int gm = tileM + wm + 16 * i + rhalf + r;
        float re = aRe[i][j][r], im = aIm[i][j][r];
        if (filt) {
          const float* fp = filt + (((long)ch * NH + gm) * NWF + gn) * 2;
          float fr = fp[0], fi = fp[1];
          float tr = re * fr - im * fi;
          im = re * fi + im * fr;
          re = tr;
        }
        Gb[(long)gm * LD + gn] = re;
        Gb[(long)gm * LD + 256 + gn] = im;
      }
    }
}

// ---------------------------------------------------------------------------
// scores per (b,c):  S[m][n] = sum_kh Gg[kh][m] * Gx[kh][n]   (complex, no conj)
// Both operands are consumed along kh, so both stage as [col][k] in LDS.
// ---------------------------------------------------------------------------
__global__ __launch_bounds__(128) void scores_kernel(
    const float* __restrict__ Gg, const float* __restrict__ Gx,
    float* __restrict__ S) {
  __shared__ float lAre[TM * PK], lAim[TM * PK];  // [m][kh]
  __shared__ float lBre[TN * PK], lBim[TN * PK];  // [n][kh]
  int bc = blockIdx.z;
  const float* Ag = Gg + (long)bc * NH * LD;
  const float* Bx = Gx + (long)bc * NH * LD;
  float* Sb = S + (long)bc * NH * LD;
  int tileN = blockIdx.x * TN, tileM = blockIdx.y * TM;
  int tid = threadIdx.x, lane = tid & 31, wave = tid >> 5;
  int wm = (wave >> 1) * 32, wn = (wave & 1) * 32;
  int lm = lane & 15, klo = (lane >> 4) * 2;
  v8f sRe[2][2] = {}, sIm[2][2] = {};
  for (int k0 = 0; k0 < NH; k0 += TK) {
    for (int i = tid; i < TK * TM; i += 128) {
      int r = i >> 6, c = i & 63;  // r: kh within tile, c: m within tile
      lAre[c * PK + r] = Ag[(long)(k0 + r) * LD + tileM + c];
      lAim[c * PK + r] = Ag[(long)(k0 + r) * LD + 256 + tileM + c];
    }
    for (int i = tid; i < TK * TN; i += 128) {
      int r = i >> 6, c = i & 63;
      lBre[c * PK + r] = Bx[(long)(k0 + r) * LD + tileN + c];
      lBim[c * PK + r] = Bx[(long)(k0 + r) * LD + 256 + tileN + c];
    }
    if (k0 + TK < NH) {
      int pk = k0 + TK + (tid >> 3);
      __builtin_prefetch(&Ag[(long)pk * LD + tileM + (tid & 7) * 8], 0, 1);
      __builtin_prefetch(&Bx[(long)pk * LD + tileN + (tid & 7) * 8], 0, 1);
    }
    __syncthreads();
    for (int kb = 0; kb < TK; kb += 4) {
      v2f are[2], aim[2], bre[2], bim[2];
      for (int i = 0; i < 2; i++) {
        int m = (wm + 16 * i + lm) * PK + kb + klo;
        are[i] = lds_pair(&lAre[m]);
        aim[i] = lds_pair(&lAim[m]);
      }
      for (int j = 0; j < 2; j++) {
        int n = (wn + 16 * j + lm) * PK + kb + klo;
        bre[j] = lds_pair(&lBre[n]);
        bim[j] = lds_pair(&lBim[n]);
      }
      for (int i = 0; i < 2; i++) {
        v2f nai; nai.x = -aim[i].x; nai.y = -aim[i].y;
        for (int j = 0; j < 2; j++) {
          sRe[i][j] = wmma4(are[i], bre[j], sRe[i][j]);
          sRe[i][j] = wmma4(nai, bim[j], sRe[i][j]);
          sIm[i][j] = wmma4(are[i], bim[j], sIm[i][j]);
          sIm[i][j] = wmma4(aim[i], bre[j], sIm[i][j]);
        }
      }
    }
    __syncthreads();
  }
  int rhalf = (lane >> 4) * 8;
  for (int i = 0; i < 2; i++)
    for (int j = 0; j < 2; j++) {
      int gn = tileN + wn + 16 * j + lm;
      for (int r = 0; r < 8; r++) {
        int gm = tileM + wm + 16 * i + rhalf + r;
        Sb[(long)gm * LD + gn] = sRe[i][j][r];
        Sb[(long)gm * LD + 256 + gn] = sIm[i][j][r];
      }
    }
}

// ---------------------------------------------------------------------------
// Per-row complex variance (biased, over kw) -> scale = rsqrt(2*pi*var)
// ---------------------------------------------------------------------------
__global__ __launch_bounds__(256) void var_scale_kernel(
    const float* __restrict__ G, float* __restrict__ scale) {
  __shared__ float sr[256], si[256], s2[256];
  long row = blockIdx.x;  // bc*256 + kh
  const float* p = G + row * LD;
  int t = threadIdx.x;
  float re = p[t], im = p[256 + t];
  sr[t] = re; si[t] = im; s2[t] = re * re + im * im;
  __syncthreads();
  for (int off = 128; off > 0; off >>= 1) {
    if (t < off) { sr[t] += sr[t + off]; si[t] += si[t + off]; s2[t] += s2[t + off]; }
    __syncthreads();
  }
  if (t == 0) {
    float mr = sr[0] * (1.0f / 256), mi = si[0] * (1.0f / 256);
    float m2 = s2[0] * (1.0f / 256);
    float var = m2 - mr * mr - mi * mi;
    scale[row] = rsqrtf(6.283185307179586f * var);
  }
}

// atten = sigmoid(scores * scale[row]) on both re and im halves, in place.
__global__ void sig_kernel(float* __restrict__ S, const float* __restrict__ scale) {
  long i = (long)blockIdx.x * 256 + threadIdx.x;  // 33554432 elements
  long row = i >> 9;
  float v = S[i] * scale[row];
  S[i] = 1.0f / (1.0f + __expf(-v));
}

// ---------------------------------------------------------------------------
// r = inv + x, then channel-first LayerNorm over C=128 (two L2-friendly passes)
// ---------------------------------------------------------------------------
__global__ __launch_bounds__(256) void ln_kernel(
    const float* __restrict__ inv, const float* __restrict__ x,
    const float* __restrict__ gamma, const float* __restrict__ beta,
    float* __restrict__ out) {
  long p = (long)blockIdx.x * 256 + threadIdx.x;  // over B*H*W
  long b = p / NHW, s = p % NHW;
  const float* pi = inv + b * (long)NC * NHW + s;
  const float* px = x + b * (long)NC * NHW + s;
  float mu = 0.0f, m2 = 0.0f;
  for (int c = 0; c < NC; c++) {
    float r = pi[(long)c * NHW] + px[(long)c * NHW];
    mu += r; m2 += r * r;
  }
  mu *= (1.0f / NC);
  m2 = m2 * (1.0f / NC) - mu * mu;
  float is = rsqrtf(m2 + 1e-6f);
  float* po = out + b * (long)NC * NHW + s;
  for (int c = 0; c < NC; c++) {
    float r = pi[(long)c * NHW] + px[(long)c * NHW];
    po[(long)c * NHW] = gamma[c] * (r - mu) * is + beta[c];
  }
}

// ---------------------------------------------------------------------------
// Host-side orchestration
// ---------------------------------------------------------------------------
extern "C" void kernel_launch(void* const* d_in, const int* in_sizes, int n_in,
                              void* d_out, int out_size, void* d_ws, size_t ws_size,
                              hipStream_t stream) {
  const float* g   = (const float*)d_in[0];
  const float* x   = (const float*)d_in[1];
  const float* wg  = (const float*)d_in[2];
  const float* bg  = (const float*)d_in[3];
  const float* wx  = (const float*)d_in[4];
  const float* bx  = (const float*)d_in[5];
  const float* fg  = (const float*)d_in[6];
  const float* fx  = (const float*)d_in[7];
  const float* lng = (const float*)d_in[8];
  const float* lnb = (const float*)d_in[9];
  float* out = (float*)d_out;

  // Workspace: 4 slots of 256*256*512 floats (128 MiB each) + tables (~3 MiB).
  const size_t SLOT = (size_t)NB * NC * NH * LD;  // 33,554,432 floats
  float* ws = (float*)d_ws;
  float* slotA = ws;                 // yg -> Gg -> inv
  float* slotB = ws + SLOT;          // yx -> Gx
  float* slotC = ws + 2 * SLOT;      // Fg -> scores/atten
  float* slotD = ws + 3 * SLOT;      // Fx -> T
  float* tabW  = ws + 4 * SLOT;                    // [510][512]
  float* tabC  = tabW + (size_t)NW * LD;           // [256][256]
  float* tabS  = tabC + (size_t)NH * NH;           // [256][256]
  float* tabIr = tabS + (size_t)NH * NH;           // [512][510]
  float* scl   = tabIr + (size_t)LD * NW;          // [65536]

  dim3 blk128(128);

  // 0) DFT basis tables
  init_dftW<<<(NW * LD + 255) / 256, 256, 0, stream>>>(tabW);
  init_dftH<<<(NH * NH + 255) / 256, 256, 0, stream>>>(tabC, tabS);
  init_irW<<<(LD * NW + 255) / 256, 256, 0, stream>>>(tabIr);

  // 1) 1x1 conv channel mixes (per-batch GEMM: [128x128]@[128x130560] + bias)
  gemm_f32_kernel<<<dim3(NHW / TN, NC / TM, NB), blk128, 0, stream>>>(
      wg, NC, 0, g, NHW, (long)NC * NHW, slotA, NHW, (long)NC * NHW,
      NC, NHW, NC, bg);
  gemm_f32_kernel<<<dim3(NHW / TN, NC / TM, NB), blk128, 0, stream>>>(
      wx, NC, 0, x, NHW, (long)NC * NHW, slotB, NHW, (long)NC * NHW,
      NC, NHW, NC, bx);

  // 2) rfft along W as real GEMM: [65536 x 510] @ [510 x 512] -> [re|im]
  gemm_f32_kernel<<<dim3(LD / TN, MROWS / TM, 1), blk128, 0, stream>>>(
      slotA, NW, 0, tabW, LD, 0, slotC, LD, 0, MROWS, LD, NW, nullptr);
  gemm_f32_kernel<<<dim3(LD / TN, MROWS / TM, 1), blk128, 0, stream>>>(
      slotB, NW, 0, tabW, LD, 0, slotD, LD, 0, MROWS, LD, NW, nullptr);

  // 3) complex DFT along H + GFNet filter epilogue
  cdft_h_kernel<<<dim3(NWF / TN, NH / TM, NB * NC), blk128, 0, stream>>>(
      slotC, slotA, tabC, tabS, fg, 0);
  cdft_h_kernel<<<dim3(NWF / TN, NH / TM, NB * NC), blk128, 0, stream>>>(
      slotD, slotB, tabC, tabS, fx, 0);

  // 4) row variance of Gg -> scale = rsqrt(2*pi*var)
  var_scale_kernel<<<NB * NC * NH, 256, 0, stream>>>(slotA, scl);

  // 5) scores = Gg^T @ Gx (complex), then complex sigmoid with per-row scale
  scores_kernel<<<dim3(NWF / TN, NWF / TM, NB * NC), blk128, 0, stream>>>(
      slotA, slotB, slotC);
  sig_kernel<<<(unsigned)(SLOT / 256), 256, 0, stream>>>(slotC, scl);

  // 6) inverse complex DFT along axis-2, then irfft along W as real GEMM
  cdft_h_kernel<<<dim3(NWF / TN, NH / TM, NB * NC), blk128, 0, stream>>>(
      slotC, slotD, tabC, tabS, nullptr, 1);
  gemm_f32_kernel<<<dim3((NW + TN - 1) / TN, MROWS / TM, 1), blk128, 0, stream>>>(
      slotD, LD, 0, tabIr, NW, 0, slotA, NW, 0, MROWS, NW, LD, nullptr);

  // 7) residual + channels-first LayerNorm
  ln_kernel<<<(NB * NHW) / 256, 256, 0, stream>>>(slotA, x, lng, lnb, out);
}